// FilterLayer_44315472560688
// MI455X (gfx1250) — compile-verified
//
#include <hip/hip_runtime.h>

// FFT-filter layer for MI455X (gfx1250):
//   h = irfft(rfft(x, ortho) * W, ortho) + x   ==  irfft(rfft(x) * (W+1))  (residual folded)
//   out = LayerNorm_H(h) * gamma + beta
// Kernel 1: per (batch, 16-channel block): pair-packed complex FFT in LDS.
//   - first 4 DIT stages + last 4 inverse DIF stages done as DFT16 matmuls via
//     v_wmma_f32_16x16x4_f32 (constant DFT fragments, data gathered from LDS)
//   - middle stages scalar radix-2 with shared twiddle table
//   - global->LDS staging via async-to-LDS when the builtin exists
// Kernel 2: LayerNorm over H=1024, one row per block, in place on d_out.

#define S_LEN   2048
#define S_HALF  1024
#define LOG2S   11
#define H_DIM   1024
#define CH_WG   16        // channels per workgroup (8 complex pairs)
#define PAIRS   8
#define TPB     256
#define PI_F    3.14159265358979323846f

typedef float v2f __attribute__((ext_vector_type(2)));
typedef float v8f __attribute__((ext_vector_type(8)));

#if defined(__HIP_DEVICE_COMPILE__)
#  if __has_builtin(__builtin_amdgcn_global_load_async_to_lds_b128) && \
      __has_builtin(__builtin_amdgcn_s_wait_asynccnt)
#    define USE_ASYNC_LDS 1
#  endif
#endif

#if defined(USE_ASYNC_LDS)
// Param 0 of the async builtin is 'int __vector(4) __device__ *' (AS1, non-const)
// per the round-1 diagnostic; mirror that type exactly.
typedef int v4i_raw __attribute__((vector_size(16)));
typedef __attribute__((address_space(1))) v4i_raw* as1_v4i;
typedef __attribute__((address_space(3))) v4i_raw* as3_v4i;
#endif

__device__ __forceinline__ float2 cmulf2(float2 a, float2 b) {
  return make_float2(a.x * b.x - a.y * b.y, a.x * b.y + a.y * b.x);
}

__global__ void __launch_bounds__(TPB)
fft_filter_kernel(const float* __restrict__ x, const float* __restrict__ wgt,
                  float* __restrict__ out) {
  __shared__ __align__(16) float smem[S_LEN * CH_WG]; // 128 KB: data[n][c], c=2p(+1)
  __shared__ __align__(16) float twds[S_HALF * 2];    // 8 KB: exp(-2*pi*i*k/S)
  float2* data = reinterpret_cast<float2*>(smem);     // data[n*PAIRS + p]
  float2* tw   = reinterpret_cast<float2*>(twds);

  const int t  = threadIdx.x;
  const int hb = (blockIdx.x & (H_DIM / CH_WG - 1)) * CH_WG;
  const int b  = blockIdx.x / (H_DIM / CH_WG);
  const size_t xbase = (size_t)b * S_LEN * H_DIM + hb;

  // ---- shared twiddle table: tw[k] = exp(-2*pi*i*k/2048), k < 1024
  for (int k = t; k < S_HALF; k += TPB) {
    float sn, cs;
    sincosf(-2.0f * PI_F * (float)k / (float)S_LEN, &sn, &cs);
    tw[k] = make_float2(cs, sn);
  }

  // ---- load x -> LDS, pair-packed complex, global bit-reversed placement
#if defined(USE_ASYNC_LDS)
  {
    const int c4 = (t & 3) * 4;                 // 16B chunk within 64B row
    for (int n = (t >> 2); n < S_LEN; n += TPB / 4) {
      const unsigned br = __brev((unsigned)n) >> (32 - LOG2S);
      const float* gp = x + xbase + (size_t)n * H_DIM + c4;
      float* lp = &smem[br * CH_WG + c4];
      __builtin_amdgcn_global_load_async_to_lds_b128((as1_v4i)gp, (as3_v4i)lp, 0, 0);
    }
    __builtin_amdgcn_s_wait_asynccnt(0);
  }
#else
  {
    const int c = t & (CH_WG - 1);
    for (int n = (t >> 4); n < S_LEN; n += TPB / CH_WG) {
      const unsigned br = __brev((unsigned)n) >> (32 - LOG2S);
      smem[br * CH_WG + c] = x[xbase + (size_t)n * H_DIM + c];
    }
  }
#endif
  __syncthreads();

  const int lane = t & 31;
  const int wv   = t >> 5;          // wave id == pair id in WMMA phases
  const int nrow = lane & 15;       // M row (A) / N column (B,C,D) index
  const int koff = (lane >> 4) * 2; // K offset for A/B fragment halves

  // ---- forward DIT stages m=1,2,4,8 == DFT16 on each 16-block (WMMA f32)
  // block bk after bit-reversed load: out[t] = sum_j u[br4(j)] * exp(-2pi i t j/16)
  {
    v2f Ar[4], Ai[4];
#pragma unroll
    for (int ks = 0; ks < 4; ++ks) {
      const int k0 = 4 * ks + koff;
      float s0, c0, s1, c1;
      sincosf(-2.0f * PI_F * (float)((nrow * k0) & 15) / 16.0f, &s0, &c0);
      sincosf(-2.0f * PI_F * (float)((nrow * (k0 + 1)) & 15) / 16.0f, &s1, &c1);
      Ar[ks][0] = c0; Ar[ks][1] = c1;
      Ai[ks][0] = s0; Ai[ks][1] = s1;
    }
    const int p = wv;
    for (int jb = 0; jb < 8; ++jb) {
      const int bk = jb * 16 + nrow;            // this lane's column = block idx
      v2f Br[4], Bi[4];
#pragma unroll
      for (int ks = 0; ks < 4; ++ks) {
        const int k0 = 4 * ks + koff;
        const int r0 = (int)(__brev((unsigned)k0) >> 28);
        const int r1 = (int)(__brev((unsigned)(k0 + 1)) >> 28);
        const float2 z0 = data[(bk * 16 + r0) * PAIRS + p];
        const float2 z1 = data[(bk * 16 + r1) * PAIRS + p];
        Br[ks][0] = z0.x; Br[ks][1] = z1.x;
        Bi[ks][0] = z0.y; Bi[ks][1] = z1.y;
      }
      v8f crr = {}, cii = {}, cim = {};
#pragma unroll
      for (int ks = 0; ks < 4; ++ks) {
        crr = __builtin_amdgcn_wmma_f32_16x16x4_f32(false, Ar[ks], false, Br[ks], (short)0, crr, false, false);
        cii = __builtin_amdgcn_wmma_f32_16x16x4_f32(false, Ai[ks], false, Bi[ks], (short)0, cii, false, false);
        cim = __builtin_amdgcn_wmma_f32_16x16x4_f32(false, Ar[ks], false, Bi[ks], (short)0, cim, false, false);
        cim = __builtin_amdgcn_wmma_f32_16x16x4_f32(false, Ai[ks], false, Br[ks], (short)0, cim, false, false);
      }
#pragma unroll
      for (int v = 0; v < 8; ++v) {
        const int mi = v + 8 * (lane >> 4);
        data[(bk * 16 + mi) * PAIRS + p] = make_float2(crr[v] - cii[v], cim[v]);
      }
    }
  }
  __syncthreads();

  // ---- forward DIT stages m=16..1024 (scalar radix-2)
  for (int ls = 4; ls <= 10; ++ls) {
    const int m = 1 << ls;
    for (int i = t; i < PAIRS * S_HALF; i += TPB) {
      const int p = i & (PAIRS - 1);
      const int q = i >> 3;
      const int j = q & (m - 1);
      const int n1 = ((q >> ls) << (ls + 1)) | j;
      const int n2 = n1 + m;
      const float2 w = tw[j << (10 - ls)];
      const float2 a = data[n1 * PAIRS + p];
      const float2 c = cmulf2(data[n2 * PAIRS + p], w);
      data[n1 * PAIRS + p] = make_float2(a.x + c.x, a.y + c.y);
      data[n2 * PAIRS + p] = make_float2(a.x - c.x, a.y - c.y);
    }
    __syncthreads();
  }

  // ---- spectral step: split packed spectra, apply (W+1), Hermitian repack.
  // Bin k and 2048-k owned exclusively by one thread -> no extra sync needed.
  {
    const float2* W2 = reinterpret_cast<const float2*>(wgt); // [1025][1024] complex
    for (int i = t; i < PAIRS * (S_HALF - 1); i += TPB) {
      const int p = i & (PAIRS - 1);
      const int k = (i >> 3) + 1;
      const float2 A = data[k * PAIRS + p];
      const float2 B = data[(S_LEN - k) * PAIRS + p];
      const float2 Xa = make_float2(0.5f * (A.x + B.x), 0.5f * (A.y - B.y));
      const float2 Xb = make_float2(0.5f * (A.y + B.y), 0.5f * (B.x - A.x));
      float2 Wa = W2[(size_t)k * H_DIM + hb + 2 * p];     Wa.x += 1.0f;
      float2 Wb = W2[(size_t)k * H_DIM + hb + 2 * p + 1]; Wb.x += 1.0f;
      const float2 Ya = cmulf2(Xa, Wa);
      const float2 Yb = cmulf2(Xb, Wb);
      data[k * PAIRS + p]           = make_float2(Ya.x - Yb.y, Ya.y + Yb.x);
      data[(S_LEN - k) * PAIRS + p] = make_float2(Ya.x + Yb.y, Yb.x - Ya.y);
    }
    if (t < PAIRS) { // DC and Nyquist: irfft keeps only Re(X*W) in these bins
      const int p = t;
      const float2 Z0 = data[p];
      const float2 ZN = data[S_HALF * PAIRS + p];
      const float2 Wa0 = W2[hb + 2 * p];
      const float2 Wb0 = W2[hb + 2 * p + 1];
      const float2 WaN = W2[(size_t)S_HALF * H_DIM + hb + 2 * p];
      const float2 WbN = W2[(size_t)S_HALF * H_DIM + hb + 2 * p + 1];
      data[p]                  = make_float2(Z0.x * (Wa0.x + 1.0f), Z0.y * (Wb0.x + 1.0f));
      data[S_HALF * PAIRS + p] = make_float2(ZN.x * (WaN.x + 1.0f), ZN.y * (WbN.x + 1.0f));
    }
  }
  __syncthreads();

  // ---- inverse DIF stages m=1024..16 (conjugate twiddles)
  for (int ls = 10; ls >= 4; --ls) {
    const int m = 1 << ls;
    for (int i = t; i < PAIRS * S_HALF; i += TPB) {
      const int p = i & (PAIRS - 1);
      const int q = i >> 3;
      const int j = q & (m - 1);
      const int n1 = ((q >> ls) << (ls + 1)) | j;
      const int n2 = n1 + m;
      float2 w = tw[j << (10 - ls)]; w.y = -w.y;
      const float2 a  = data[n1 * PAIRS + p];
      const float2 bb = data[n2 * PAIRS + p];
      data[n1 * PAIRS + p] = make_float2(a.x + bb.x, a.y + bb.y);
      data[n2 * PAIRS + p] = cmulf2(make_float2(a.x - bb.x, a.y - bb.y), w);
    }
    __syncthreads();
  }

  // ---- inverse DIF stages m=8,4,2,1 == per-16-block matrix via WMMA:
  // out[j] = sum_t w[t] * exp(+2pi i t br4(j)/16)
  {
    const int rb = (int)(__brev((unsigned)nrow) >> 28);
    v2f Ar[4], Ai[4];
#pragma unroll
    for (int ks = 0; ks < 4; ++ks) {
      const int k0 = 4 * ks + koff;
      float s0, c0, s1, c1;
      sincosf(2.0f * PI_F * (float)((rb * k0) & 15) / 16.0f, &s0, &c0);
      sincosf(2.0f * PI_F * (float)((rb * (k0 + 1)) & 15) / 16.0f, &s1, &c1);
      Ar[ks][0] = c0; Ar[ks][1] = c1;
      Ai[ks][0] = s0; Ai[ks][1] = s1;
    }
    const int p = wv;
    for (int jb = 0; jb < 8; ++jb) {
      const int bk = jb * 16 + nrow;
      v2f Br[4], Bi[4];
#pragma unroll
      for (int ks = 0; ks < 4; ++ks) {
        const int k0 = 4 * ks + koff;
        const float2 z0 = data[(bk * 16 + k0) * PAIRS + p];
        const float2 z1 = data[(bk * 16 + k0 + 1) * PAIRS + p];
        Br[ks][0] = z0.x; Br[ks][1] = z1.x;
        Bi[ks][0] = z0.y; Bi[ks][1] = z1.y;
      }
      v8f crr = {}, cii = {}, cim = {};
#pragma unroll
      for (int ks = 0; ks < 4; ++ks) {
        crr = __builtin_amdgcn_wmma_f32_16x16x4_f32(false, Ar[ks], false, Br[ks], (short)0, crr, false, false);
        cii = __builtin_amdgcn_wmma_f32_16x16x4_f32(false, Ai[ks], false, Bi[ks], (short)0, cii, false, false);
        cim = __builtin_amdgcn_wmma_f32_16x16x4_f32(false, Ar[ks], false, Bi[ks], (short)0, cim, false, false);
        cim = __builtin_amdgcn_wmma_f32_16x16x4_f32(false, Ai[ks], false, Br[ks], (short)0, cim, false, false);
      }
#pragma unroll
      for (int v = 0; v < 8; ++v) {
        const int mi = v + 8 * (lane >> 4);
        data[(bk * 16 + mi) * PAIRS + p] = make_float2(crr[v] - cii[v], cim[v]);
      }
    }
  }
  __syncthreads();

  // ---- store h[n] = data[br11(n)] / 2048 (bit-reversed readout), coalesced
  {
    const int c = t & (CH_WG - 1);
    const float inv = 1.0f / (float)S_LEN;
    for (int n = (t >> 4); n < S_LEN; n += TPB / CH_WG) {
      const unsigned br = __brev((unsigned)n) >> (32 - LOG2S);
      out[xbase + (size_t)n * H_DIM + c] = smem[br * CH_WG + c] * inv;
    }
  }
}

// ---- LayerNorm over H=1024, eps=1e-12, biased variance; in place on d_out.
__global__ void __launch_bounds__(256)
ln_kernel(float* __restrict__ io, const float* __restrict__ gamma,
          const float* __restrict__ beta) {
  __shared__ float red[8];
  const size_t row = (size_t)blockIdx.x;
  float4* rp = reinterpret_cast<float4*>(io + row * H_DIM);
  const int t = threadIdx.x;
  const float4 v = rp[t];

  float s = v.x + v.y + v.z + v.w;
  for (int off = 16; off > 0; off >>= 1) s += __shfl_down(s, off, 32);
  if ((t & 31) == 0) red[t >> 5] = s;
  __syncthreads();
  float mean = 0.f;
  for (int i = 0; i < 8; ++i) mean += red[i];
  mean *= (1.0f / (float)H_DIM);
  __syncthreads();

  const float dx = v.x - mean, dy = v.y - mean, dz = v.z - mean, dw = v.w - mean;
  float q = dx * dx + dy * dy + dz * dz + dw * dw;
  for (int off = 16; off > 0; off >>= 1) q += __shfl_down(q, off, 32);
  if ((t & 31) == 0) red[t >> 5] = q;
  __syncthreads();
  float var = 0.f;
  for (int i = 0; i < 8; ++i) var += red[i];
  var *= (1.0f / (float)H_DIM);

  const float rs = rsqrtf(var + 1e-12f);
  const float4 g  = reinterpret_cast<const float4*>(gamma)[t];
  const float4 bb = reinterpret_cast<const float4*>(beta)[t];
  float4 o;
  o.x = dx * rs * g.x + bb.x;
  o.y = dy * rs * g.y + bb.y;
  o.z = dz * rs * g.z + bb.z;
  o.w = dw * rs * g.w + bb.w;
  rp[t] = o;
}

extern "C" void kernel_launch(void* const* d_in, const int* in_sizes, int n_in,
                              void* d_out, int out_size, void* d_ws, size_t ws_size,
                              hipStream_t stream) {
  (void)n_in; (void)out_size; (void)d_ws; (void)ws_size;
  const float* x     = (const float*)d_in[0];
  const float* w     = (const float*)d_in[1];
  const float* gamma = (const float*)d_in[2];
  const float* beta  = (const float*)d_in[3];
  float* out = (float*)d_out;

  const int B = in_sizes[0] / (S_LEN * H_DIM);
  dim3 g1((unsigned)(B * (H_DIM / CH_WG)));   // 32 * 64 = 2048 workgroups
  fft_filter_kernel<<<g1, TPB, 0, stream>>>(x, w, out);

  dim3 g2((unsigned)(B * S_LEN));             // 65536 rows
  ln_kernel<<<g2, 256, 0, stream>>>(out, gamma, beta);
}